// CutoutColor_23519240913599
// MI455X (gfx1250) — compile-verified
//
#include <hip/hip_runtime.h>

// Native 16-byte vector type (clang ext vector: supports .x/.y/.z/.w and
// works with __builtin_nontemporal_store, unlike HIP's struct float4).
typedef float v4f __attribute__((ext_vector_type(4)));

#define NBATCH 4096
#define NCH    9
#define HH     84
#define WW     84
#define PATCH  28

constexpr int ELEMS   = NBATCH * NCH * HH * WW;   // 260,112,384 fp32
constexpr int NVEC    = ELEMS / 4;                // 65,028,096 vec4 (exact)
constexpr int CHUNK   = 256;                      // vec4 per block-iteration
constexpr int NCHUNKS = (NVEC + CHUNK - 1) / CHUNK; // 254,016 (exact)
constexpr int PLANE_V = (HH * WW) / 4;            // 1764 vec4 per (n,c) plane
constexpr int ROW_V   = WW / 4;                   // 21 vec4 per row

__global__ __launch_bounds__(CHUNK) void CutoutColor_kernel(
    const v4f*  __restrict__ x,       // [N*C*H*W/4]
    const float* __restrict__ colors, // [N*9]  (index == plane id)
    const int*  __restrict__ tops,    // [N]
    const int*  __restrict__ lefts,   // [N]
    v4f*        __restrict__ out)     // [N*C*H*W/4]
{
    // Double-buffered LDS staging: 2 x 256 x 16B = 8KB (of 320KB/WGP).
    __shared__ v4f buf[2][CHUNK];
    const int tid = threadIdx.x;

    // Low 32 bits of a generic pointer into LDS == LDS byte address
    // (flat-aperture rule).
    const unsigned ldsAddr0 = (unsigned)(size_t)&buf[0][tid];
    const unsigned ldsAddr1 = (unsigned)(size_t)&buf[1][tid];

    int chunk        = blockIdx.x;
    const int stride = gridDim.x;

    // ---- pipeline prologue: stage first chunk into buffer 0 (ASYNCcnt++) ----
    {
        int v = chunk * CHUNK + tid;
        if (v < NVEC) {
            asm volatile("global_load_async_to_lds_b128 %0, %1, off"
                         :: "v"(ldsAddr0), "v"(x + v) : "memory");
        }
    }

    int cur = 0;
    while (chunk < NCHUNKS) {
        const int nextChunk = chunk + stride;   // uniform across the block

        if (nextChunk < NCHUNKS) {
            // Prefetch next chunk into the other buffer, then wait until at
            // most ONE async op is outstanding: async loads complete in
            // order, so the current chunk's data is now in LDS.
            int vn = nextChunk * CHUNK + tid;
            unsigned ldsNext = (cur == 0) ? ldsAddr1 : ldsAddr0;
            if (vn < NVEC) {
                asm volatile("global_load_async_to_lds_b128 %0, %1, off"
                             :: "v"(ldsNext), "v"(x + vn) : "memory");
            }
            asm volatile("s_wait_asynccnt 0x1" ::: "memory");
        } else {
            asm volatile("s_wait_asynccnt 0x0" ::: "memory");
        }

        // ---- process current chunk out of LDS ----
        int v = chunk * CHUNK + tid;
        if (v < NVEC) {
            v4f xv = buf[cur][tid];             // ds_load_b128 (own slot; no barrier)

            int p  = v / PLANE_V;               // plane id = n*9 + c
            int r  = v - p * PLANE_V;           // vec4 within plane
            int h  = r / ROW_V;                 // row
            int c0 = (r - h * ROW_V) << 2;      // first column of this vec4
            int n  = p / NCH;                   // sample

            int  top = tops[n];
            v4f  o   = xv;
            if ((unsigned)(h - top) < (unsigned)PATCH) {   // row inside patch
                int   left = lefts[n];
                float col  = colors[p];         // colors flat-indexed by plane
                int d = c0 - left;              // per-component column test
                if ((unsigned)(d)     < (unsigned)PATCH) o.x = col;
                if ((unsigned)(d + 1) < (unsigned)PATCH) o.y = col;
                if ((unsigned)(d + 2) < (unsigned)PATCH) o.z = col;
                if ((unsigned)(d + 3) < (unsigned)PATCH) o.w = col;
            }
            // Output is write-once, never re-read: non-temporal B128 store.
            __builtin_nontemporal_store(o, out + v);
        }

        cur  ^= 1;
        chunk = nextChunk;
    }
}

extern "C" void kernel_launch(void* const* d_in, const int* in_sizes, int n_in,
                              void* d_out, int out_size, void* d_ws, size_t ws_size,
                              hipStream_t stream) {
    (void)in_sizes; (void)n_in; (void)out_size; (void)d_ws; (void)ws_size;
    const v4f*   x      = (const v4f*)  d_in[0];
    const float* colors = (const float*)d_in[1];
    const int*   tops   = (const int*)  d_in[2];
    const int*   lefts  = (const int*)  d_in[3];
    v4f* out = (v4f*)d_out;

    // Persistent-ish grid: ~31 chunks per block -> the 2-deep async pipeline
    // amortizes; 8192 blocks x 8 wave32 keeps every WGP saturated.
    int grid = 8192;
    if (grid > NCHUNKS) grid = NCHUNKS;
    CutoutColor_kernel<<<grid, CHUNK, 0, stream>>>(x, colors, tops, lefts, out);
}